// GATConv4_22454089024233
// MI455X (gfx1250) — compile-verified
//
#include <hip/hip_runtime.h>
#include <math.h>

// Problem constants (from reference)
#define N_NODES 2708
#define IN_CH   1433
#define IN_CH4  1432           // largest multiple of 4 <= IN_CH
#define OUT_CH  64
#define NP      2720           // N padded to multiple of 16 (and of 32)
#define MT      (NP / 16)      // 170 16-tiles
#define NT32    (NP / 32)      // 85  32-tiles

typedef __attribute__((ext_vector_type(2))) float v2f;
typedef __attribute__((ext_vector_type(8))) float v8f;

// Native CDNA5 fp32 WMMA: D(16x16) = A(16x4) * B(4x16) + C
__device__ __forceinline__ v8f wmma_f32(v2f a, v2f b, v8f c) {
    return __builtin_amdgcn_wmma_f32_16x16x4_f32(false, a, false, b,
                                                 (short)0, c, false, false);
}

__device__ __forceinline__ v2f ldv2(const float* p) {   // 8B-aligned pair load
    return *(const v2f*)p;
}

// ---------------------------------------------------------------------------
// 1) K/Q/V = X @ W^T   (W is [64,1433] row-major, so B[k][n] = W[n][k])
//    grid = (MT, 3), block = 128 (4 waves, one 16-col n-tile each)
// ---------------------------------------------------------------------------
__global__ __launch_bounds__(128)
void proj_kernel(const float* __restrict__ X,
                 const float* __restrict__ Wk,
                 const float* __restrict__ Wq,
                 const float* __restrict__ Wv,
                 float* __restrict__ Kp,
                 float* __restrict__ Qp,
                 float* __restrict__ Vp)
{
    const int wave = threadIdx.x >> 5;
    const int lane = threadIdx.x & 31;
    const int half = lane >> 4;
    const int l16  = lane & 15;
    const int mb = blockIdx.x * 16;
    const int nb = wave * 16;

    const float* W;
    float* O;
    if (blockIdx.y == 0)      { W = Wk; O = Kp; }
    else if (blockIdx.y == 1) { W = Wq; O = Qp; }
    else                      { W = Wv; O = Vp; }

    const int m  = mb + l16;
    const int mc = (m < N_NODES) ? m : (N_NODES - 1);   // clamped A row
    const int n  = nb + l16;                            // always < 64

    const float* xrow = X + (size_t)mc * IN_CH;
    const float* wrow = W + (size_t)n  * IN_CH;

    v8f acc = {0.f,0.f,0.f,0.f,0.f,0.f,0.f,0.f};
    for (int k0 = 0; k0 < IN_CH4; k0 += 4) {
        const int ka = k0 + 2 * half;
        v2f a, b;
        a.x = xrow[ka];     a.y = xrow[ka + 1];
        b.x = wrow[ka];     b.y = wrow[ka + 1];
        acc = wmma_f32(a, b, acc);
    }
    {   // branch-free tail: only k = 1432 is valid, and only for half == 0
        const float km = half ? 0.f : 1.f;
        v2f a, b;
        a.x = xrow[IN_CH4] * km;  a.y = 0.f;
        b.x = wrow[IN_CH4] * km;  b.y = 0.f;
        acc = wmma_f32(a, b, acc);
    }
#pragma unroll
    for (int r = 0; r < 8; ++r) {
        const int mm = mb + r + 8 * half;               // C layout: M = r + 8*half
        O[mm * OUT_CH + n] = (mm < N_NODES) ? acc[r] : 0.f;
    }
}

// ---------------------------------------------------------------------------
// 2) E = (Q @ K^T) / 64     grid = (MT, MT), block = 32
// ---------------------------------------------------------------------------
__global__ __launch_bounds__(32)
void qk_kernel(const float* __restrict__ Qp,
               const float* __restrict__ Kp,
               float* __restrict__ E)
{
    const int lane = threadIdx.x & 31;
    const int half = lane >> 4;
    const int l16  = lane & 15;
    const int mb = blockIdx.x * 16;
    const int nb = blockIdx.y * 16;
    const int m = mb + l16;
    const int n = nb + l16;

    const float* qrow = Qp + (size_t)m * OUT_CH;
    const float* krow = Kp + (size_t)n * OUT_CH;

    v8f acc = {0.f,0.f,0.f,0.f,0.f,0.f,0.f,0.f};
#pragma unroll
    for (int k0 = 0; k0 < OUT_CH; k0 += 4) {
        const int ka = k0 + 2 * half;                   // even -> 8B aligned
        acc = wmma_f32(ldv2(qrow + ka), ldv2(krow + ka), acc);
    }
    const float s = 1.0f / 64.0f;
#pragma unroll
    for (int r = 0; r < 8; ++r) {
        const int mm = mb + r + 8 * half;
        E[mm * NP + n] = acc[r] * s;
    }
}

// ---------------------------------------------------------------------------
// 3) T = lmbd ⊙ (U^T @ E)    32x32 register tile per wave (2x2 WMMA tiles)
//    grid = (NT32, NT32), block = 32
// ---------------------------------------------------------------------------
__global__ __launch_bounds__(32)
void ute_kernel(const float* __restrict__ U,
                const float* __restrict__ E,
                const float* __restrict__ lmbd,
                float* __restrict__ T)
{
    const int lane = threadIdx.x & 31;
    const int half = lane >> 4;
    const int l16  = lane & 15;
    const int mb = blockIdx.x * 32;
    const int nb = blockIdx.y * 32;

    const int m0  = mb + l16;
    const int m1  = mb + 16 + l16;
    const int mc0 = (m0 < N_NODES) ? m0 : (N_NODES - 1);  // clamped U columns
    const int mc1 = (m1 < N_NODES) ? m1 : (N_NODES - 1);
    const int n0  = nb + l16;                             // < NP always
    const int n1  = nb + 16 + l16;

    v8f a00 = {0.f,0.f,0.f,0.f,0.f,0.f,0.f,0.f};
    v8f a01 = a00, a10 = a00, a11 = a00;

    for (int k0 = 0; k0 < N_NODES; k0 += 4) {             // 2708 % 4 == 0
        const int ka = k0 + 2 * half;
        const int kp = (ka + 32 < N_NODES) ? ka + 32 : N_NODES - 1;
        __builtin_prefetch(&E[kp * NP + n0], 0, 1);
        const float* u0 = U + (size_t)ka * N_NODES;       // A[m][k] = U[k][m]
        const float* u1 = u0 + N_NODES;
        const float* e0 = E + (size_t)ka * NP;
        const float* e1 = e0 + NP;
        v2f fa0, fa1, fb0, fb1;
        fa0.x = u0[mc0];  fa0.y = u1[mc0];
        fa1.x = u0[mc1];  fa1.y = u1[mc1];
        fb0.x = e0[n0];   fb0.y = e1[n0];
        fb1.x = e0[n1];   fb1.y = e1[n1];
        a00 = wmma_f32(fa0, fb0, a00);
        a01 = wmma_f32(fa0, fb1, a01);
        a10 = wmma_f32(fa1, fb0, a10);
        a11 = wmma_f32(fa1, fb1, a11);
    }

#pragma unroll
    for (int r = 0; r < 8; ++r) {
        const int ma  = mb + r + 8 * half;                // tiles 0x
        const int mbb = ma + 16;                          // tiles 1x
        const int mac = (ma  < N_NODES) ? ma  : (N_NODES - 1);
        const int mbc = (mbb < N_NODES) ? mbb : (N_NODES - 1);
        const float ka_m = (ma  < N_NODES) ? 1.f : 0.f;
        const float kb_m = (mbb < N_NODES) ? 1.f : 0.f;
        const float la = lmbd[mac] * ka_m;
        const float lb = lmbd[mbc] * kb_m;
        T[ma  * NP + n0] = a00[r] * la;
        T[ma  * NP + n1] = a01[r] * la;
        T[mbb * NP + n0] = a10[r] * lb;
        T[mbb * NP + n1] = a11[r] * lb;
    }
}

// ---------------------------------------------------------------------------
// 4) E2 = |U @ T|    32x32 register tile per wave (2x2 WMMA tiles)
//    grid = (NT32, NT32), block = 32
// ---------------------------------------------------------------------------
__global__ __launch_bounds__(32)
void ut2_kernel(const float* __restrict__ U,
                const float* __restrict__ T,
                float* __restrict__ E2)
{
    const int lane = threadIdx.x & 31;
    const int half = lane >> 4;
    const int l16  = lane & 15;
    const int mb = blockIdx.x * 32;
    const int nb = blockIdx.y * 32;

    const int m0  = mb + l16;
    const int m1  = mb + 16 + l16;
    const int mc0 = (m0 < N_NODES) ? m0 : (N_NODES - 1);
    const int mc1 = (m1 < N_NODES) ? m1 : (N_NODES - 1);
    const int n0  = nb + l16;
    const int n1  = nb + 16 + l16;

    const float* ur0 = U + (size_t)mc0 * N_NODES;         // rows: 2708 floats, 16B aligned
    const float* ur1 = U + (size_t)mc1 * N_NODES;

    v8f a00 = {0.f,0.f,0.f,0.f,0.f,0.f,0.f,0.f};
    v8f a01 = a00, a10 = a00, a11 = a00;

    for (int k0 = 0; k0 < N_NODES; k0 += 4) {
        const int ka = k0 + 2 * half;                     // even -> 8B aligned
        const int kp = (ka + 32 < N_NODES) ? ka + 32 : N_NODES - 1;
        __builtin_prefetch(&T[kp * NP + n0], 0, 1);
        const float* t0 = T + (size_t)ka * NP;
        const float* t1 = t0 + NP;
        v2f fb0, fb1;
        fb0.x = t0[n0];  fb0.y = t1[n0];
        fb1.x = t0[n1];  fb1.y = t1[n1];
        const v2f fa0 = ldv2(ur0 + ka);
        const v2f fa1 = ldv2(ur1 + ka);
        a00 = wmma_f32(fa0, fb0, a00);
        a01 = wmma_f32(fa0, fb1, a01);
        a10 = wmma_f32(fa1, fb0, a10);
        a11 = wmma_f32(fa1, fb1, a11);
    }

#pragma unroll
    for (int r = 0; r < 8; ++r) {
        const int ma  = mb + r + 8 * half;
        const int mbb = ma + 16;
        const float ka_m = (ma  < N_NODES) ? 1.f : 0.f;
        const float kb_m = (mbb < N_NODES) ? 1.f : 0.f;
        E2[ma  * NP + n0] = fabsf(a00[r]) * ka_m;
        E2[ma  * NP + n1] = fabsf(a01[r]) * ka_m;
        E2[mbb * NP + n0] = fabsf(a10[r]) * kb_m;
        E2[mbb * NP + n1] = fabsf(a11[r]) * kb_m;
    }
}

// ---------------------------------------------------------------------------
// 5) masked softmax per row (in place over E buffer); pad cols zeroed
//    grid = N, block = 256
// ---------------------------------------------------------------------------
__global__ __launch_bounds__(256)
void softmax_kernel(const int* __restrict__ A, float* __restrict__ E)
{
    __shared__ float red[256];
    const int m   = blockIdx.x;
    const int tid = threadIdx.x;
    float*     e    = E + (size_t)m * NP;
    const int* arow = A + (size_t)m * N_NODES;

    float mx = -INFINITY;
    for (int j = tid; j < N_NODES; j += 256)
        if (arow[j] != 0) mx = fmaxf(mx, e[j]);
    red[tid] = mx;
    __syncthreads();
    for (int s = 128; s > 0; s >>= 1) {
        if (tid < s) red[tid] = fmaxf(red[tid], red[tid + s]);
        __syncthreads();
    }
    mx = red[0];
    __syncthreads();

    float sum = 0.f;
    for (int j = tid; j < N_NODES; j += 256)
        if (arow[j] != 0) sum += expf(e[j] - mx);
    red[tid] = sum;
    __syncthreads();
    for (int s = 128; s > 0; s >>= 1) {
        if (tid < s) red[tid] += red[tid + s];
        __syncthreads();
    }
    sum = red[0];

    const float inv = 1.0f / sum;
    for (int j = tid; j < NP; j += 256) {
        float v = 0.f;
        if (j < N_NODES && arow[j] != 0) v = expf(e[j] - mx) * inv;
        e[j] = v;
    }
}

// ---------------------------------------------------------------------------
// 6) H = alpha @ V   grid = MT, block = 128 (4 waves over the 4 n-tiles)
// ---------------------------------------------------------------------------
__global__ __launch_bounds__(128)
void av_kernel(const float* __restrict__ alpha,
               const float* __restrict__ Vp,
               float* __restrict__ H)
{
    const int wave = threadIdx.x >> 5;
    const int lane = threadIdx.x & 31;
    const int half = lane >> 4;
    const int l16  = lane & 15;
    const int mb = blockIdx.x * 16;
    const int nb = wave * 16;
    const int m = mb + l16;
    const int n = nb + l16;

    const float* arow = alpha + (size_t)m * NP;           // 2720 floats/row: 16B aligned

    v8f acc = {0.f,0.f,0.f,0.f,0.f,0.f,0.f,0.f};
#pragma unroll 2
    for (int k0 = 0; k0 < NP; k0 += 4) {                  // pad region is zero
        const int ka = k0 + 2 * half;                     // even -> 8B aligned
        v2f b;
        b.x = Vp[ka * OUT_CH + n];
        b.y = Vp[(ka + 1) * OUT_CH + n];
        acc = wmma_f32(ldv2(arow + ka), b, acc);
    }
#pragma unroll
    for (int r = 0; r < 8; ++r) {
        const int mm = mb + r + 8 * half;
        if (mm < N_NODES) H[mm * OUT_CH + n] = acc[r];
    }
}

// ---------------------------------------------------------------------------
extern "C" void kernel_launch(void* const* d_in, const int* in_sizes, int n_in,
                              void* d_out, int out_size, void* d_ws, size_t ws_size,
                              hipStream_t stream)
{
    const float* X    = (const float*)d_in[0];
    const int*   A    = (const int*)  d_in[1];
    const float* U    = (const float*)d_in[2];
    const float* Wk   = (const float*)d_in[3];
    const float* Wq   = (const float*)d_in[4];
    const float* Wv   = (const float*)d_in[5];
    const float* lmbd = (const float*)d_in[6];
    float*       H    = (float*)d_out;

    char* ws = (char*)d_ws;
    size_t off = 0;
    float* Kp   = (float*)(ws + off); off += (size_t)NP * OUT_CH * sizeof(float);
    float* Qp   = (float*)(ws + off); off += (size_t)NP * OUT_CH * sizeof(float);
    float* Vp   = (float*)(ws + off); off += (size_t)NP * OUT_CH * sizeof(float);
    float* Ebuf = (float*)(ws + off); off += (size_t)NP * NP * sizeof(float);
    float* Tbuf = (float*)(ws + off); off += (size_t)NP * NP * sizeof(float);
    (void)ws_size; (void)in_sizes; (void)n_in; (void)out_size;

    proj_kernel   <<<dim3(MT, 3),      128, 0, stream>>>(X, Wk, Wq, Wv, Kp, Qp, Vp);
    qk_kernel     <<<dim3(MT, MT),      32, 0, stream>>>(Qp, Kp, Ebuf);
    ute_kernel    <<<dim3(NT32, NT32),  32, 0, stream>>>(U, Ebuf, lmbd, Tbuf);
    ut2_kernel    <<<dim3(NT32, NT32),  32, 0, stream>>>(U, Tbuf, Ebuf);
    softmax_kernel<<<N_NODES,          256, 0, stream>>>(A, Ebuf);
    av_kernel     <<<MT,               128, 0, stream>>>(Ebuf, Vp, H);
}